// EdgeConvBlock_20770461843673
// MI455X (gfx1250) — compile-verified
//
#include <hip/hip_runtime.h>

typedef float v2f __attribute__((ext_vector_type(2)));
typedef float v8f __attribute__((ext_vector_type(8)));
typedef unsigned int v4u __attribute__((ext_vector_type(4)));
typedef unsigned int v8u __attribute__((ext_vector_type(8)));

#define B_   8
#define C_   64
#define N_   4096
#define K_   20
#define O_   64
#define CNT_ ((float)(B_ * N_ * K_))
#define EPS_ 1e-5f
#define INF_ (__builtin_inff())

// ---------------------------------------------------------------- sq[b][n] = sum_c x^2
__global__ void sq_kernel(const float* __restrict__ x, float* __restrict__ sq) {
  int t = blockIdx.x * blockDim.x + threadIdx.x;   // over B*N
  int b = t >> 12, n = t & (N_ - 1);
  const float* xp = x + (size_t)b * C_ * N_ + n;
  float s = 0.f;
#pragma unroll
  for (int c = 0; c < C_; ++c) { float v = xp[c * N_]; s += v * v; }
  sq[t] = s;
}

// ------------------------------------------- W2[r][c]: r<64 -> W1-W2 ; r>=64 -> W2
__global__ void w2_kernel(const float* __restrict__ W, float* __restrict__ W2) {
  int r = threadIdx.x;                              // 0..127
  for (int c = 0; c < C_; ++c) {
    float v;
    if (r < O_) v = W[r * 128 + c] - W[r * 128 + 64 + c];
    else        v = W[(r - 64) * 128 + 64 + c];
    W2[r * C_ + c] = v;
  }
}

// ---------------------------------------------------------------- distances + top-20
// One WG (4 waves) per (batch, 16-row block). Row-block operand (64x16 f32 tile,
// row stride N) staged into LDS by the Tensor Data Mover. Each wave sweeps 1/4 of
// the 256 column tiles; G tile = 16x f32 WMMA (K=4 each, C=64). Output mapping:
// M = column, N = row, so lane (lane&15) owns one row; per-lane top-20 list in LDS.
__launch_bounds__(128)
__global__ void dist_topk_kernel(const float* __restrict__ x,
                                 const float* __restrict__ sq,
                                 int* __restrict__ knn) {
  __shared__ float xrow[C_ * 16];     // row-block operand, channel-major
  __shared__ float sqrow[16];
  __shared__ float ldist[128 * K_];
  __shared__ int   lidx[128 * K_];

  const int tid  = threadIdx.x;
  const int wave = tid >> 5, lane = tid & 31;
  const int wg   = blockIdx.x;
  const int b    = wg >> 8;                 // 256 row blocks per batch
  const int rowbase = (wg & 255) << 4;
  const float* xb = x + (size_t)b * C_ * N_;

  if (wave == 0) {
    // ---- TDM: 2D tile load, tensor = x[b] (dim0=N fastest, dim1=C rows),
    //      tile 16 x 64, packed rows -> xrow[c*16 + r]
    unsigned long long ga = (unsigned long long)(uintptr_t)(xb + rowbase);
    unsigned int lds_off  = (unsigned int)(uintptr_t)xrow;   // low 32b of LDS aperture addr
    v4u g0; v8u g1;
    g0[0] = 1u;                                              // count=1 (valid D#)
    g0[1] = lds_off;                                         // lds_addr
    g0[2] = (unsigned int)ga;                                // global_addr[31:0]
    g0[3] = ((unsigned int)(ga >> 32) & 0x01FFFFFFu) | 0x80000000u; // addr[56:32] | type=2
    g1[0] = 0x00020000u;                                     // data_size=2 (4 bytes)
    g1[1] = ((unsigned int)N_ & 0xFFFFu) << 16;              // tensor_dim0[15:0]
    g1[2] = ((unsigned int)C_ << 16) | ((unsigned int)N_ >> 16); // dim1[15:0] | dim0[31:16]
    g1[3] = 16u << 16;                                       // tile_dim0 = 16
    g1[4] = (unsigned int)C_;                                // tile_dim1 = 64 (tile_dim2=0)
    g1[5] = (unsigned int)N_;                                // tensor_dim0_stride[31:0]
    g1[6] = 0u;
    g1[7] = 0u;
    asm volatile("tensor_load_to_lds %0, %1" :: "s"(g0), "s"(g1) : "memory");
    __builtin_amdgcn_s_wait_tensorcnt(0);
  }
  if (tid < 16) sqrow[tid] = sq[b * N_ + rowbase + tid];
  const int lbase = tid * K_;
#pragma unroll
  for (int i = 0; i < K_; ++i) { ldist[lbase + i] = INF_; lidx[lbase + i] = 0; }
  __syncthreads();

  const int   lm   = lane & 15;             // row within block / col-lane of A
  const int   half = lane >> 4;
  const float rowsq = sqrow[lm];
  const float* xrp = xrow + half * 32 + lm; // B operand base (LDS, imm offsets)
  const float* sqb = sq + b * N_;
  float curmax = INF_;
  int   curpos = 0;

  for (int ct = wave; ct < N_ / 16; ct += 4) {
    const int colbase = ct << 4;
    const float* colp = xb + (size_t)(half * 2) * N_ + colbase + lm; // A base (imm offsets)
    if (ct + 4 < N_ / 16) __builtin_prefetch(colp + 64, 0, 0);       // next tile of this wave

    v8f acc = {};
#pragma unroll
    for (int s = 0; s < 16; ++s) {
      v2f a, bb;
      a.x  = colp[(size_t)(s * 4)     * N_];      // A: M=column, k-slots by half-wave
      a.y  = colp[(size_t)(s * 4 + 1) * N_];
      bb.x = xrp[s * 64];                         // B: N=row (LDS)
      bb.y = xrp[s * 64 + 16];
      acc = __builtin_amdgcn_wmma_f32_16x16x4_f32(false, a, false, bb,
                                                  (short)0, acc, false, false);
    }

    const float* sqp = sqb + colbase + half * 8;  // 32B-aligned
    float4 sa = *(const float4*)sqp;
    float4 sb4 = *(const float4*)(sqp + 4);
    float sqv[8] = {sa.x, sa.y, sa.z, sa.w, sb4.x, sb4.y, sb4.z, sb4.w};
#pragma unroll
    for (int j = 0; j < 8; ++j) {
      const float d = sqv[j] + rowsq - 2.0f * acc[j];
      const int col = colbase + half * 8 + j;     // D: VGPR j -> M=j (+8 upper half)
      if (d < curmax) {
        ldist[lbase + curpos] = d;
        lidx [lbase + curpos] = col;
        float m = -INF_; int p = 0;
#pragma unroll
        for (int i = 0; i < K_; ++i) {
          float v = ldist[lbase + i];
          if (v > m) { m = v; p = i; }
        }
        curmax = m; curpos = p;
      }
    }
  }
  __syncthreads();

  // merge the 8 partial lists of each row (2 half-waves x 4 waves)
  if (tid < 16) {
    const int r = tid;
    const int out_base = ((b << 12) + rowbase + r) * K_;
    for (int ksel = 0; ksel < K_; ++ksel) {
      float best = INF_; int bslot = 0; int bidx = 0;
      for (int s = 0; s < 8; ++s) {
        int src = (s >> 1) * 32 + ((s & 1) << 4) + r;
        for (int i = 0; i < K_; ++i) {
          float v = ldist[src * K_ + i];
          if (v < best) { best = v; bslot = src * K_ + i; bidx = lidx[src * K_ + i]; }
        }
      }
      ldist[bslot] = INF_;
      knn[out_base + ksel] = bidx;
    }
  }
}

// ------------------------------------------------- uvT[b][n][0:128] = W2 @ x  (M=r, N=n)
__launch_bounds__(128)
__global__ void uv_kernel(const float* __restrict__ x, const float* __restrict__ W2,
                          float* __restrict__ uvT) {
  const int tid  = threadIdx.x;
  const int wave = tid >> 5, lane = tid & 31;
  const int wg   = blockIdx.x;              // B * 8 rtiles * 64 ntile-groups
  const int b    = wg >> 9;
  const int rt   = (wg >> 6) & 7;
  const int nt   = ((wg & 63) << 2) + wave;
  const int rbase = rt << 4, nbase = nt << 4;
  const int lm = lane & 15, half = lane >> 4;
  const float* xb = x + (size_t)b * C_ * N_;

  const float* w2p = W2 + (rbase + lm) * C_ + half * 2;          // contiguous pairs
  const float* bxp = xb + (size_t)(half * 2) * N_ + nbase + lm;  // imm channel offsets

  v8f acc = {};
#pragma unroll
  for (int s = 0; s < 16; ++s) {
    v2f a = *(const v2f*)(w2p + s * 4);
    v2f bb;
    bb.x = bxp[(size_t)(s * 4)     * N_];
    bb.y = bxp[(size_t)(s * 4 + 1) * N_];
    acc = __builtin_amdgcn_wmma_f32_16x16x4_f32(false, a, false, bb,
                                                (short)0, acc, false, false);
  }
  const int n = nbase + lm;
  float4 f0 = make_float4(acc[0], acc[1], acc[2], acc[3]);
  float4 f1 = make_float4(acc[4], acc[5], acc[6], acc[7]);
  float* dst = uvT + ((size_t)((b << 12) + n)) * 128 + rbase + half * 8;
  *(float4*)(dst)     = f0;
  *(float4*)(dst + 4) = f1;
}

// ------------- gather pass: y = u + v[knn]; per-(b,o,n) max/min over k + BN partials
__launch_bounds__(256)
__global__ void stats_kernel(const float* __restrict__ uvT, const int* __restrict__ knn,
                             float* __restrict__ ymax, float* __restrict__ ymin,
                             float* __restrict__ part) {
  __shared__ float ps[8][128];              // [wave][ 0..63 sum | 64..127 sumsq ]
  const int tid  = threadIdx.x;
  const int wave = tid >> 5, lane = tid & 31;
  const int p = blockIdx.x * 8 + wave;      // (b,n) pair
  const int b = p >> 12, n = p & (N_ - 1);
  const int o2 = lane * 2;

  const float2 u2 = *(const float2*)(uvT + ((size_t)((b << 12) + n)) * 128 + o2);
  const int* kp = knn + ((size_t)((b << 12) + n)) * K_;

  float mx0 = -INF_, mx1 = -INF_, mn0 = INF_, mn1 = INF_;
  float s0 = 0.f, s1 = 0.f, q0 = 0.f, q1 = 0.f;
#pragma unroll 4
  for (int k = 0; k < K_; ++k) {
    int idx = kp[k];
    const float2 v2v = *(const float2*)(uvT + ((size_t)((b << 12) + idx)) * 128 + 64 + o2);
    float y0 = u2.x + v2v.x;
    float y1 = u2.y + v2v.y;
    mx0 = fmaxf(mx0, y0); mn0 = fminf(mn0, y0); s0 += y0; q0 += y0 * y0;
    mx1 = fmaxf(mx1, y1); mn1 = fminf(mn1, y1); s1 += y1; q1 += y1 * y1;
  }
  *(float2*)(ymax + ((size_t)((b << 12) + n)) * 64 + o2) = make_float2(mx0, mx1);
  *(float2*)(ymin + ((size_t)((b << 12) + n)) * 64 + o2) = make_float2(mn0, mn1);

  ps[wave][o2]          = s0;  ps[wave][o2 + 1]      = s1;
  ps[wave][64 + o2]     = q0;  ps[wave][64 + o2 + 1] = q1;
  __syncthreads();
  if (tid < 128) {
    float t = 0.f;
    for (int w = 0; w < 8; ++w) t += ps[w][tid];   // fixed order -> deterministic
    part[(size_t)blockIdx.x * 128 + tid] = t;
  }
}

// ------------------------------------------------------------- deterministic reduce
__global__ void reduce_kernel(const float* __restrict__ part, float* __restrict__ totals) {
  __shared__ float red[256];
  const int q = blockIdx.x;                 // 0..127
  const int t = threadIdx.x;
  float s = 0.f;
  for (int i = t; i < 4096; i += 256) s += part[(size_t)i * 128 + q];
  red[t] = s;
  __syncthreads();
  for (int st = 128; st > 0; st >>= 1) {
    if (t < st) red[t] += red[t + st];
    __syncthreads();
  }
  if (t == 0) totals[q] = red[0];
}

// ------------------------------------------------------------- BN scale/shift
__global__ void bnparam_kernel(const float* __restrict__ totals,
                               const float* __restrict__ gamma,
                               const float* __restrict__ beta,
                               float* __restrict__ ss) {
  int o = threadIdx.x;
  if (o < O_) {
    float mean = totals[o] / CNT_;
    float var  = totals[64 + o] / CNT_ - mean * mean;
    float inv  = rsqrtf(var + EPS_);
    float sc   = gamma[o] * inv;
    ss[o]      = sc;
    ss[64 + o] = beta[o] - mean * sc;
  }
}

// --------------------------------- out[b][o][n] = relu(sc * (sc>=0 ? ymax : ymin) + sh)
__global__ void final_kernel(const float* __restrict__ ymax, const float* __restrict__ ymin,
                             const float* __restrict__ ss, float* __restrict__ out) {
  int t = blockIdx.x * blockDim.x + threadIdx.x;   // B*64*N, n fastest
  int n = t & (N_ - 1);
  int o = (t >> 12) & 63;
  int b = t >> 18;
  float sc = ss[o], sh = ss[64 + o];
  size_t src = ((size_t)((b << 12) + n)) * 64 + o;
  float y = (sc >= 0.f) ? ymax[src] : ymin[src];
  float r = sc * y + sh;
  out[t] = r > 0.f ? r : 0.f;
}

extern "C" void kernel_launch(void* const* d_in, const int* in_sizes, int n_in,
                              void* d_out, int out_size, void* d_ws, size_t ws_size,
                              hipStream_t stream) {
  const float* x     = (const float*)d_in[0];
  const float* W     = (const float*)d_in[1];
  const float* gamma = (const float*)d_in[2];
  const float* beta  = (const float*)d_in[3];
  float* out = (float*)d_out;

  char* w = (char*)d_ws;
  auto alloc = [&](size_t bytes) -> char* {
    char* p = w;
    w += (bytes + 255) & ~(size_t)255;
    return p;
  };
  float* sq     = (float*)alloc((size_t)B_ * N_ * 4);              // 128 KB
  float* W2     = (float*)alloc((size_t)128 * C_ * 4);             // 32 KB
  int*   knn    = (int*)  alloc((size_t)B_ * N_ * K_ * 4);         // 2.5 MB
  float* uvT    = (float*)alloc((size_t)B_ * N_ * 128 * 4);        // 16 MB
  float* ymax   = (float*)alloc((size_t)B_ * N_ * 64 * 4);         // 8 MB
  float* ymin   = (float*)alloc((size_t)B_ * N_ * 64 * 4);         // 8 MB
  float* part   = (float*)alloc((size_t)4096 * 128 * 4);           // 2 MB
  float* totals = (float*)alloc(128 * 4);
  float* ssb    = (float*)alloc(128 * 4);

  sq_kernel       <<<B_ * N_ / 256, 256, 0, stream>>>(x, sq);
  w2_kernel       <<<1, 128, 0, stream>>>(W, W2);
  dist_topk_kernel<<<B_ * (N_ / 16), 128, 0, stream>>>(x, sq, knn);
  uv_kernel       <<<B_ * 8 * 64, 128, 0, stream>>>(x, W2, uvT);
  stats_kernel    <<<B_ * N_ / 8, 256, 0, stream>>>(uvT, knn, ymax, ymin, part);
  reduce_kernel   <<<128, 256, 0, stream>>>(part, totals);
  bnparam_kernel  <<<1, 64, 0, stream>>>(totals, gamma, beta, ssb);
  final_kernel    <<<B_ * 64 * N_ / 256, 256, 0, stream>>>(ymax, ymin, ssb, out);
}